// FilteredTopKModel_50886772523396
// MI455X (gfx1250) — compile-verified
//
#include <hip/hip_runtime.h>
#include <stdint.h>

// Emit final indices via GLOBAL_STORE_ASYNC_FROM_LDS_B32 (ASYNCcnt path).
// Flip to 0 if the toolchain's assembler rejects the async mnemonics.
#define USE_ASYNC_EMIT 1

namespace {

constexpr int kRowsC   = 512;
constexpr int kColsC   = 131072;
constexpr int kNT      = 1024;                 // 32 wave32 waves -> fills one WGP
constexpr int kVec     = 4;
constexpr int kChunk   = kNT * kVec;           // 4096 elements per block-iteration
constexpr int kIters   = kColsC / kChunk;      // 32
constexpr int kMaxBins = 4096;                 // 12-bit digits
constexpr int kListCap = 4096;                 // candidate list (32 KB LDS)

struct SMem {
  uint32_t hist[kMaxBins];                     // 16 KB
  uint32_t part[64];
  uint32_t ctrl[8];                            // 0:digit 1:kRem 3:equalCnt 4:listCnt 5:fallbackTaken
  unsigned long long list[kListCap];           // 32 KB (key<<32 | idx)
  uint32_t scan[kNT];                          // 4 KB (emit staging / fallback scan)
};

// Monotone map: float order == unsigned order of the key.
__device__ __forceinline__ uint32_t monokey(float f) {
  uint32_t u = __float_as_uint(f);
  return u ^ ((uint32_t)((int32_t)u >> 31) | 0x80000000u);
}

// After a histogram round: find digit containing rank ctrl[1], update ctrl.
__device__ void select_digit(SMem& sm, int t, int nbins) {
  const int per = nbins >> 6;                  // 64 groups
  if (t < 64) {
    uint32_t s = 0;
    for (int j = 0; j < per; ++j) s += sm.hist[t * per + j];
    sm.part[t] = s;
  }
  __syncthreads();
  if (t == 0) {
    uint32_t kRem  = sm.ctrl[1];
    uint32_t below = 0;
    int g = 0;
    while (g < 63 && below + sm.part[g] < kRem) { below += sm.part[g]; ++g; }
    int d = g * per;
    while (d < nbins - 1 && below + sm.hist[d] < kRem) { below += sm.hist[d]; ++d; }
    sm.ctrl[0] = (uint32_t)d;
    sm.ctrl[1] = kRem - below;                 // rank within chosen bin (1-based)
    sm.ctrl[3] = sm.hist[d];                   // population of chosen bin
  }
  __syncthreads();
}

template <bool kUseWs>
__device__ __forceinline__ void load_quad(const float4* xr4, const int4* fr4,
                                          const uint4* wr4, int gi, int fval,
                                          float z, float shim, uint32_t key[4]) {
  if (kUseWs) {
    uint4 kv = wr4[gi];
    key[0] = kv.x; key[1] = kv.y; key[2] = kv.z; key[3] = kv.w;
  } else {
    float4 xv = xr4[gi];
    int4   fc = fr4[gi];
    key[0] = monokey(xv.x + (fc.x == fval ? z : shim));
    key[1] = monokey(xv.y + (fc.y == fval ? z : shim));
    key[2] = monokey(xv.z + (fc.z == fval ? z : shim));
    key[3] = monokey(xv.w + (fc.w == fval ? z : shim));
  }
}

template <bool kUseWs>
__global__ __launch_bounds__(kNT) void filtered_topk_kernel(
    const float* __restrict__ x, const int* __restrict__ filt,
    const int* __restrict__ fvalP, const float* __restrict__ zP,
    const float* __restrict__ shimP, int K, int* __restrict__ out,
    uint32_t* __restrict__ wsKeys) {
  __shared__ SMem sm;
  const int row  = (int)blockIdx.x;
  const int t    = (int)threadIdx.x;
  const int   fval = fvalP[0];
  const float z    = zP[0];
  const float shim = shimP[0];

  const float*    xrow = x    + (size_t)row * kColsC;
  const int*      frow = filt + (size_t)row * kColsC;
  uint32_t*       wrow = kUseWs ? (wsKeys + (size_t)row * kColsC) : nullptr;
  const float4*   xr4  = reinterpret_cast<const float4*>(xrow);
  const int4*     fr4  = reinterpret_cast<const int4*>(frow);
  uint4*          wr4  = reinterpret_cast<uint4*>(wrow);

  // ---------- pass 0: compute keys (cache to ws), histogram bits [31:20] ----------
  for (int b = t; b < kMaxBins; b += kNT) sm.hist[b] = 0;
  if (t == 0) sm.ctrl[1] = (uint32_t)K;
  __syncthreads();
  for (int i = 0; i < kIters; ++i) {
    int gi = i * kNT + t;
    if (i + 1 < kIters) {                      // -> global_prefetch_b8
      __builtin_prefetch(&xr4[gi + kNT], 0, 0);
      __builtin_prefetch(&fr4[gi + kNT], 0, 0);
    }
    float4 xv = xr4[gi];
    int4   fc = fr4[gi];
    uint32_t k0 = monokey(xv.x + (fc.x == fval ? z : shim));
    uint32_t k1 = monokey(xv.y + (fc.y == fval ? z : shim));
    uint32_t k2 = monokey(xv.z + (fc.z == fval ? z : shim));
    uint32_t k3 = monokey(xv.w + (fc.w == fval ? z : shim));
    if (kUseWs) wr4[gi] = make_uint4(k0, k1, k2, k3);
    atomicAdd(&sm.hist[k0 >> 20], 1u);
    atomicAdd(&sm.hist[k1 >> 20], 1u);
    atomicAdd(&sm.hist[k2 >> 20], 1u);
    atomicAdd(&sm.hist[k3 >> 20], 1u);
  }
  __syncthreads();
  select_digit(sm, t, kMaxBins);
  uint32_t prefix = sm.ctrl[0];                // 12 bits known

  // ---------- round 1: histogram bits [19:8] among prefix matches ----------
  for (int b = t; b < kMaxBins; b += kNT) sm.hist[b] = 0;
  __syncthreads();
  for (int i = 0; i < kIters; ++i) {
    int gi = i * kNT + t;
    uint32_t key[4];
    load_quad<kUseWs>(xr4, fr4, wr4, gi, fval, z, shim, key);
#pragma unroll
    for (int j = 0; j < 4; ++j)
      if ((key[j] >> 20) == prefix) atomicAdd(&sm.hist[(key[j] >> 8) & 0xFFFu], 1u);
  }
  __syncthreads();
  select_digit(sm, t, kMaxBins);
  prefix = (prefix << 12) | sm.ctrl[0];        // 24 bits known

  // ---------- round 2: histogram bits [7:0] ----------
  for (int b = t; b < 256; b += kNT) sm.hist[b] = 0;
  __syncthreads();
  for (int i = 0; i < kIters; ++i) {
    int gi = i * kNT + t;
    uint32_t key[4];
    load_quad<kUseWs>(xr4, fr4, wr4, gi, fval, z, shim, key);
#pragma unroll
    for (int j = 0; j < 4; ++j)
      if ((key[j] >> 8) == prefix) atomicAdd(&sm.hist[key[j] & 0xFFu], 1u);
  }
  __syncthreads();
  select_digit(sm, t, 256);
  const uint32_t kth = (prefix << 8) | sm.ctrl[0];   // exact k-th smallest key
  const uint32_t T   = sm.ctrl[1];                   // # equal-key elems to keep
  const uint32_t E   = sm.ctrl[3];                   // # elems equal to kth
  const uint32_t L   = (uint32_t)K - T;              // # elems strictly below kth
  const bool fits    = (L + E) <= (uint32_t)kListCap;

  // ---------- collect candidates ----------
  for (int b = t; b < kListCap; b += kNT) sm.list[b] = ~0ull;
  if (t == 0) sm.ctrl[4] = 0;
  __syncthreads();
  for (int i = 0; i < kIters; ++i) {
    int gi = i * kNT + t;
    uint32_t key[4];
    load_quad<kUseWs>(xr4, fr4, wr4, gi, fval, z, shim, key);
#pragma unroll
    for (int j = 0; j < 4; ++j) {
      uint32_t kj = key[j];
      if (kj < kth || (fits && kj == kth)) {
        uint32_t p = atomicAdd(&sm.ctrl[4], 1u);
        if (p < (uint32_t)kListCap)
          sm.list[p] = ((unsigned long long)kj << 32) | (uint32_t)(gi * 4 + j);
      }
    }
  }
  __syncthreads();

  // ---------- bitonic sort ascending by (key, idx) ----------
  for (unsigned size = 2; size <= (unsigned)kListCap; size <<= 1) {
    for (unsigned stride = size >> 1; stride > 0; stride >>= 1) {
      for (int i = t; i < kListCap; i += kNT) {
        int p = i ^ (int)stride;
        if (p > i) {
          bool up = (((unsigned)i & size) == 0u);
          unsigned long long a = sm.list[i], b = sm.list[p];
          if (up ? (a > b) : (a < b)) { sm.list[i] = b; sm.list[p] = a; }
        }
      }
      __syncthreads();
    }
  }

  // ---------- emit indices ----------
  // Stage the winning indices in LDS, then DMA them out with the CDNA5 async
  // LDS->global store path (ASYNCcnt), draining with s_wait_asynccnt.
  int* orow = out + (size_t)row * K;
  const uint32_t outCount = fits ? (uint32_t)K : L;
  for (uint32_t base = 0; base < outCount; base += (uint32_t)kNT) {
    uint32_t r = base + (uint32_t)t;
    if (r < outCount) sm.scan[t] = (uint32_t)(sm.list[r] & 0xFFFFFFFFull);
    __syncthreads();                           // drains DScnt: staging visible
    if (r < outCount) {
#if USE_ASYNC_EMIT
      unsigned long long gaddr = (unsigned long long)(uintptr_t)(orow + r);
      // Low 32 bits of a generic LDS pointer == workgroup-relative LDS byte
      // address (ISA 10.2 aperture layout).
      uint32_t ldsAddr = (uint32_t)(uintptr_t)&sm.scan[t];
      asm volatile("global_store_async_from_lds_b32 %0, %1, off"
                   :: "v"(gaddr), "v"(ldsAddr) : "memory");
#else
      orow[r] = (int)sm.scan[t];
#endif
    }
#if USE_ASYNC_EMIT
    asm volatile("s_wait_asynccnt 0x0" ::: "memory");  // staging reusable
#endif
    __syncthreads();
  }

  if (!fits) {
    // Deterministic fallback for massive ties at the threshold: take the first
    // T equal-key indices in ascending index order via block-wide scans.
    if (t == 0) sm.ctrl[5] = 0;
    __syncthreads();
    const int nChunks = kColsC / kNT;
    for (int c = 0; c < nChunks; ++c) {
      uint32_t taken = sm.ctrl[5];
      if (taken >= T) break;                   // uniform condition
      int gi = c * kNT + t;
      uint32_t key;
      if (kUseWs) {
        key = wrow[gi];
      } else {
        float xv = xrow[gi];
        int   fc = frow[gi];
        key = monokey(xv + (fc == fval ? z : shim));
      }
      uint32_t m = (key == kth) ? 1u : 0u;
      sm.scan[t] = m;
      __syncthreads();
      for (int off = 1; off < kNT; off <<= 1) {   // Hillis-Steele inclusive scan
        uint32_t v   = sm.scan[t];
        uint32_t add = (t >= off) ? sm.scan[t - off] : 0u;
        __syncthreads();
        sm.scan[t] = v + add;
        __syncthreads();
      }
      uint32_t incl = sm.scan[t];
      uint32_t excl = incl - m;
      if (m && taken + excl < T) orow[L + taken + excl] = gi;
      uint32_t total = sm.scan[kNT - 1];
      __syncthreads();
      if (t == 0) sm.ctrl[5] = taken + total;
      __syncthreads();
    }
  }
}

}  // namespace

extern "C" void kernel_launch(void* const* d_in, const int* in_sizes, int n_in,
                              void* d_out, int out_size, void* d_ws, size_t ws_size,
                              hipStream_t stream) {
  (void)in_sizes; (void)n_in;
  const float* x     = (const float*)d_in[0];
  const int*   filt  = (const int*)d_in[1];
  const int*   fval  = (const int*)d_in[2];
  const float* z     = (const float*)d_in[3];
  const float* shim  = (const float*)d_in[4];
  // d_in[5] holds k on device; derive the same value host-side for sizing.
  const int K = out_size / kRowsC;             // 100 for the reference shapes
  int* out = (int*)d_out;

  const size_t wsNeed = (size_t)kRowsC * (size_t)kColsC * sizeof(uint32_t);
  dim3 grid(kRowsC), block(kNT);
  if (ws_size >= wsNeed) {
    filtered_topk_kernel<true><<<grid, block, 0, stream>>>(
        x, filt, fval, z, shim, K, out, (uint32_t*)d_ws);
  } else {
    filtered_topk_kernel<false><<<grid, block, 0, stream>>>(
        x, filt, fval, z, shim, K, out, nullptr);
  }
}